// TensorProduct_46497315947093
// MI455X (gfx1250) — compile-verified
//
#include <hip/hip_runtime.h>

// CDNA5 / gfx1250: channelwise irrep tensor product as a tall-skinny fp32 GEMM
//   out(Z*128, 9) = Y(Z*128, 81) x W(81, 9),  Y[row, i*9+j] = x1[row,i]*x2[row,j]
// mapped onto V_WMMA_F32_16X16X4_F32 (K padded 81->84 => 21 WMMAs per 16-row tile,
// N padded 9->16). v3 improvements:
//   - three independent WMMA accumulator chains (3-deep D->C issue spacing)
//   - per-g store indices as independent 32-bit v_mad => all stores in
//     saddr + scale_offset form (no 64-bit vector address chains)

typedef float v2f __attribute__((ext_vector_type(2)));
typedef float v8f __attribute__((ext_vector_type(8)));

#define TP_MUL   128
#define TP_OP    1152      // 128*(1+3+5)
#define TP_KPAD  84        // 81 padded to multiple of 4
#define TP_NCOL  16        // 9 padded to 16
#define TP_NCOEF 615

__global__ __launch_bounds__(256)
void tp_wmma_kernel(const float* __restrict__ x1,
                    const float* __restrict__ x2,
                    const float* __restrict__ coeffs,
                    float* __restrict__ out,
                    int Z)
{
    // ---- dense padded coefficient matrix W[k][n] in LDS (84x16 f32 = 5.25 KB) ----
    __shared__ float W[TP_KPAD * TP_NCOL];

    const int tid = threadIdx.x;
    for (int i = tid; i < TP_KPAD * TP_NCOL; i += 256) W[i] = 0.0f;
    __syncthreads();

    // Path table (l1,l2,l3) in reference order, and flat coeff offsets.
    static constexpr int L1[15]  = {0,0,0, 1,1,1,1,1,1, 2,2,2,2,2,2};
    static constexpr int L2[15]  = {0,1,2, 0,1,1,1,2,2, 0,1,1,2,2,2};
    static constexpr int L3[15]  = {0,1,2, 1,0,1,2,1,2, 2,1,2,0,1,2};
    static constexpr int COFF[16] = {0,1,10,35,44,53,80,125,170,245,270,315,390,415,490,615};
    static constexpr int IOFF[3] = {0,1,4};   // offset of irrep l within the 9-dim vector

    if (tid < TP_NCOEF) {
        int pid = 0;
        while (COFF[pid + 1] <= tid) pid++;
        const int l1 = L1[pid], l2 = L2[pid], l3 = L3[pid];
        const int d2 = 2*l2 + 1, d3 = 2*l3 + 1;
        int loc = tid - COFF[pid];
        const int k = loc % d3;
        const int j = (loc / d3) % d2;
        const int i = loc / (d3 * d2);
        const int gi = IOFF[l1] + i;     // 0..8
        const int gj = IOFF[l2] + j;     // 0..8
        const int gk = IOFF[l3] + k;     // 0..8
        W[(gi * 9 + gj) * TP_NCOL + gk] = coeffs[tid];
    }
    __syncthreads();

    const int lane = tid & 31;
    const int wave = tid >> 5;
    const int half = lane >> 4;    // 0: K+{0,1}, 1: K+{2,3} within each 16x16x4 step
    const int nn   = lane & 15;    // output column / row-within-tile

    // ---- B fragments held in VGPRs for the whole kernel (42 VGPRs) ----
    // f32 B 4x16 layout (mirrors A): VGPR0 = row K (lanes 0-15) / K+2 (lanes 16-31),
    // VGPR1 = K+1 / K+3, column N = lane%16.
    v2f bfrag[21];
    #pragma unroll
    for (int t = 0; t < 21; ++t) {
        const int kk = 4 * t + 2 * half;
        bfrag[t].x = W[(kk    ) * TP_NCOL + nn];
        bfrag[t].y = W[(kk + 1) * TP_NCOL + nn];
    }

    // Per-lane output-address decomposition for column nn (segments 1/3/5 wide).
    const bool valid = (nn < 9);
    int sbase, sstr, sm;
    if (nn == 0)     { sbase = 0;   sstr = 1; sm = 0;      }
    else if (nn < 4) { sbase = 128; sstr = 3; sm = nn - 1; }
    else             { sbase = 512; sstr = 5; sm = nn - 4; }

    const int NT = Z * (TP_MUL / 16);          // 16-row tiles; 8 tiles per z
    const int gw = blockIdx.x * 8 + wave;
    const int nw = gridDim.x * 8;

    for (int tile = gw; tile < NT; tile += nw) {
        // tile is wave-uniform by construction: force the z-block base scalar so
        // all global accesses become saddr + 32-bit voffset (+imm) form.
        const int zb = __builtin_amdgcn_readfirstlane(tile >> 3) * TP_OP;
        const int u0 = (tile & 7) << 4;        // first channel of this tile
        const int r  = u0 + nn;                // row this lane loads (dup across halves)
        const float* p1 = x1 + zb;             // SGPR base
        const float* p2 = x2 + zb;             // SGPR base

        // Prefetch next tile's z-block (each lane pulls a distinct 144B slice).
        const int tnext = tile + nw;
        if (tnext < NT) {
            const int zb2 = __builtin_amdgcn_readfirstlane(tnext >> 3) * TP_OP;
            __builtin_prefetch(x1 + zb2 + lane * 36, 0, 3);
            __builtin_prefetch(x2 + zb2 + lane * 36, 0, 3);
        }

        // Gather the 9 irrep components of x1/x2 for row r (segmented layout).
        const int r3 = r * 3, r5 = r * 5;
        float a1[9], a2[9];
        a1[0] = p1[r];
        a2[0] = p2[r];
        #pragma unroll
        for (int m = 0; m < 3; ++m) {
            a1[1 + m] = p1[128 + r3 + m];      // imm offset (128+m)*4
            a2[1 + m] = p2[128 + r3 + m];
        }
        #pragma unroll
        for (int m = 0; m < 5; ++m) {
            a1[4 + m] = p1[512 + r5 + m];      // imm offset (512+m)*4
            a2[4 + m] = p2[512 + r5 + m];
        }

        // ---- A fragments: Y[r, k] = x1[r, k/9] * x2[r, k%9], zero-padded k>=81 ----
        v2f afrag[21];
        if (half == 0) {
            #pragma unroll
            for (int t = 0; t < 21; ++t) {
                const int k0 = 4 * t, k1 = 4 * t + 1;
                afrag[t].x = (k0 < 81) ? a1[k0 / 9] * a2[k0 % 9] : 0.0f;
                afrag[t].y = (k1 < 81) ? a1[k1 / 9] * a2[k1 % 9] : 0.0f;
            }
        } else {
            #pragma unroll
            for (int t = 0; t < 21; ++t) {
                const int k0 = 4 * t + 2, k1 = 4 * t + 3;
                afrag[t].x = (k0 < 81) ? a1[k0 / 9] * a2[k0 % 9] : 0.0f;
                afrag[t].y = (k1 < 81) ? a1[k1 / 9] * a2[k1 % 9] : 0.0f;
            }
        }

        // ---- 21x V_WMMA_F32_16X16X4_F32 in three independent accumulator chains ----
        v8f c0 = {}, c1 = {}, c2 = {};
        #pragma unroll
        for (int t = 0; t < 21; t += 3) {
            c0 = __builtin_amdgcn_wmma_f32_16x16x4_f32(
                    false, afrag[t    ], false, bfrag[t    ], (short)0, c0, false, false);
            c1 = __builtin_amdgcn_wmma_f32_16x16x4_f32(
                    false, afrag[t + 1], false, bfrag[t + 1], (short)0, c1, false, false);
            c2 = __builtin_amdgcn_wmma_f32_16x16x4_f32(
                    false, afrag[t + 2], false, bfrag[t + 2], (short)0, c2, false, false);
        }

        // ---- scatter store: lane holds column nn, rows g + 8*half of the tile ----
        if (valid) {
            float* po = out + zb;                                  // SGPR base
            const int idx0 = sbase + sm + (u0 + 8 * half) * sstr;
            #pragma unroll
            for (int g = 0; g < 8; ++g) {
                // independent 32-bit index per store => saddr + scale_offset form
                const unsigned idx = (unsigned)(idx0 + g * sstr);
                __builtin_nontemporal_store((c0[g] + c1[g]) + c2[g], po + (size_t)idx);
            }
        }
    }
}

extern "C" void kernel_launch(void* const* d_in, const int* in_sizes, int n_in,
                              void* d_out, int out_size, void* d_ws, size_t ws_size,
                              hipStream_t stream) {
    const float* x1 = (const float*)d_in[0];
    const float* x2 = (const float*)d_in[1];
    const float* cf = (const float*)d_in[2];
    float* out = (float*)d_out;

    const int Z  = in_sizes[0] / TP_OP;   // 50000
    const int nt = Z * (TP_MUL / 16);     // 400000 tiles
    int blocks = (nt + 7) / 8;            // 8 waves (tiles/iter) per 256-thread block
    if (blocks > 4096) blocks = 4096;     // persistent grid-stride over tiles
    tp_wmma_kernel<<<blocks, 256, 0, stream>>>(x1, x2, cf, out, Z);
}